// FusedCombineBmm_46119358824617
// MI455X (gfx1250) — compile-verified
//
#include <hip/hip_runtime.h>
#include <math.h>

typedef __attribute__((ext_vector_type(2))) float v2f;
typedef __attribute__((ext_vector_type(8))) float v8f;

#define T_EXP 8
#define M_TOT 16384
#define K_TOT 128
#define N_TOT 1024

#define MBLK 128        // block tile rows
#define NBLK 64         // block tile cols
#define LDSN 72         // padded LDS row stride (dwords): 2*72 % 64 = 16 -> conflict-free half-wave split

// Branch-free fp32 erf (njuffa minimax, max err ~2.5 ulp), large-path exp via
// a single v_exp_f32 (argument is always <= -1.2, safely in range).
__device__ __forceinline__ float erf_branchfree(float a) {
    const float t = __builtin_fabsf(a);
    const float s = a * a;
    // |a| < 0.921875: polynomial in a^2
    float p = -5.96761703e-4f;
    p = __builtin_fmaf(p, s,  4.99119423e-3f);
    p = __builtin_fmaf(p, s, -2.67681349e-2f);
    p = __builtin_fmaf(p, s,  1.12819925e-1f);
    p = __builtin_fmaf(p, s, -3.76125336e-1f);
    p = __builtin_fmaf(p, s,  1.28379166e-1f);
    p = __builtin_fmaf(p, a, a);
    // |a| >= 0.921875: 1 - exp(poly(t))
    float q = __builtin_fmaf(-1.72853470e-5f, t, 3.83197126e-4f);
    const float u = __builtin_fmaf(-3.88396438e-3f, t, 2.42546219e-2f);
    q = __builtin_fmaf(q, s, u);
    q = __builtin_fmaf(q, t, -1.06777877e-1f);
    q = __builtin_fmaf(q, t, -6.34846687e-1f);
    q = __builtin_fmaf(q, t, -1.28717512e-1f);
    q = __builtin_fmaf(q, t, -t);
    q = 1.0f - __builtin_amdgcn_exp2f(q * 1.44269504088896340736f); // exp(q)
    q = __builtin_copysignf(q, a);
    return (t > 0.921875f) ? q : p;
}

__device__ __forceinline__ float gelu_exact(float x) {
    return 0.5f * x * (1.0f + erf_branchfree(x * 0.70710678118654752f));
}

__global__ __launch_bounds__(256)
void fused_bmm_bias_gelu_f32(const float* __restrict__ A,
                             const float* __restrict__ W,
                             const float* __restrict__ bias,
                             float* __restrict__ out) {
    __shared__ float wlds[K_TOT * LDSN];

    const int t  = blockIdx.z;
    const size_t m0 = (size_t)blockIdx.y * MBLK;
    const int n0 = blockIdx.x * NBLK;

    const float* __restrict__ Ap = A    + (size_t)t * M_TOT * K_TOT;
    const float* __restrict__ Wp = W    + (size_t)t * K_TOT * N_TOT;
    const float* __restrict__ bp = bias + (size_t)t * N_TOT;
    float* __restrict__       Op = out  + (size_t)t * M_TOT * N_TOT;

    // ---- stage W tile [K_TOT x NBLK] into padded LDS (float4 transfers) ----
    {
        const int tid = threadIdx.x;
        #pragma unroll
        for (int i = 0; i < (K_TOT * (NBLK / 4)) / 256; ++i) {
            const int idx = tid + i * 256;
            const int kr  = idx / (NBLK / 4);
            const int c4  = (idx % (NBLK / 4)) * 4;
            const float4 w = *reinterpret_cast<const float4*>(Wp + (size_t)kr * N_TOT + n0 + c4);
            *reinterpret_cast<float4*>(&wlds[kr * LDSN + c4]) = w;
        }
    }
    __syncthreads();

    const int lane = threadIdx.x & 31;
    const int wave = threadIdx.x >> 5;
    const int nw   = wave & 3;        // 4 waves across N
    const int mw   = wave >> 2;       // 2 waves across M
    const int half = lane >> 4;       // lane-half: K offset +2 for A/B fragments
    const int l16  = lane & 15;

    const int    nc    = nw * 16 + l16;            // column within block tile
    const size_t mbase = m0 + (size_t)mw * 64;     // 4 x 16-row subtiles per wave

    v8f c0 = {}, c1 = {}, c2 = {}, c3 = {};

    for (int kb = 0; kb < K_TOT; kb += 4) {
        const int kk = kb + half * 2;

        // B fragment 4x16: vgpr0 -> row kk, vgpr1 -> row kk+1 (per lane-half)
        v2f b;
        b.x = wlds[kk * LDSN + nc];
        b.y = wlds[(kk + 1) * LDSN + nc];

        // A fragments 16x4: per-lane contiguous k-pair -> 8B global loads
        const v2f a0 = *reinterpret_cast<const v2f*>(Ap + (mbase +  0 + l16) * K_TOT + kk);
        const v2f a1 = *reinterpret_cast<const v2f*>(Ap + (mbase + 16 + l16) * K_TOT + kk);
        const v2f a2 = *reinterpret_cast<const v2f*>(Ap + (mbase + 32 + l16) * K_TOT + kk);
        const v2f a3 = *reinterpret_cast<const v2f*>(Ap + (mbase + 48 + l16) * K_TOT + kk);

        // 8-arg form: (neg_a, A, neg_b, B, c_mod, C, reuse_a, reuse_b)
        c0 = __builtin_amdgcn_wmma_f32_16x16x4_f32(false, a0, false, b, (short)0, c0, false, false);
        c1 = __builtin_amdgcn_wmma_f32_16x16x4_f32(false, a1, false, b, (short)0, c1, false, false);
        c2 = __builtin_amdgcn_wmma_f32_16x16x4_f32(false, a2, false, b, (short)0, c2, false, false);
        c3 = __builtin_amdgcn_wmma_f32_16x16x4_f32(false, a3, false, b, (short)0, c3, false, false);
    }

    // ---- epilogue: bias + exact GELU (branch-free), non-temporal stores ----
    const int   gcol = n0 + nc;
    const float bv   = bp[gcol];

    #pragma unroll
    for (int i = 0; i < 4; ++i) {
        const v8f& c = (i == 0) ? c0 : (i == 1) ? c1 : (i == 2) ? c2 : c3;
        #pragma unroll
        for (int j = 0; j < 8; ++j) {
            const size_t row = mbase + (size_t)i * 16 + half * 8 + j;
            const float  g   = gelu_exact(c[j] + bv);
            // output is write-once streaming (512 MiB): NT store keeps L2 for A/W
            __builtin_nontemporal_store(g, &Op[row * N_TOT + gcol]);
        }
    }
}

extern "C" void kernel_launch(void* const* d_in, const int* in_sizes, int n_in,
                              void* d_out, int out_size, void* d_ws, size_t ws_size,
                              hipStream_t stream) {
    const float* A    = (const float*)d_in[0];  // [T,1,M,K]
    const float* W    = (const float*)d_in[1];  // [T,1,K,N]
    const float* bias = (const float*)d_in[2];  // [T,1,1,N]
    float*       out  = (float*)d_out;          // [T,1,M,N]

    dim3 grid(N_TOT / NBLK, M_TOT / MBLK, T_EXP); // (16, 128, 8)
    fused_bmm_bias_gelu_f32<<<grid, 256, 0, stream>>>(A, W, bias, out);
}